// TMSCNNAveragePoolingGPU_74294344286541
// MI455X (gfx1250) — compile-verified
//
#include <hip/hip_runtime.h>

typedef __attribute__((ext_vector_type(2))) float v2f;
typedef __attribute__((ext_vector_type(8))) float v8f;

#define NFILT 64          // F
#define WAVES_PER_BLOCK 8 // 256 threads, wave32

// One wave32 per target segment.
// Segment-sum via V_WMMA_F32_16X16X4_F32 with A = ones(16x4):
//   D = ones * B + C  => every row of D = column-sums of B (+C), exact f32.
// B tile k-rows: lanes 0-15 carry gathered rows e+0 (b.x) / e+1 (b.y),
// lanes 16-31 carry rows e+2 (b.x) / e+3 (b.y).  4 chunks of 16 cols cover F=64.
// All segment-walk state is forced into SGPRs via readfirstlane so the loop is
// straight-line scalar control flow (EXEC stays all-1s for the WMMAs).
__global__ __launch_bounds__(256)
void seg_mean_wmma_kernel(const float* __restrict__ src,
                          const int*   __restrict__ idx,
                          const int*   __restrict__ spans,
                          float*       __restrict__ out,
                          int n_tgt)
{
    const int lane = threadIdx.x & 31;
    // wave id is uniform per wave -- make that explicit for the compiler
    const int wave = __builtin_amdgcn_readfirstlane((int)(threadIdx.x >> 5));
    const int t = __builtin_amdgcn_readfirstlane((int)blockIdx.x) * WAVES_PER_BLOCK + wave;
    if (t >= n_tgt) return;                    // uniform branch

    const int beg = __builtin_amdgcn_readfirstlane(spans[t]);
    const int end = __builtin_amdgcn_readfirstlane(spans[t + 1]);
    const int cnt = end - beg;
    const int elast = (cnt > 0) ? end - 1 : beg;   // uniform clamp anchor

    const int  col = lane & 15;
    const bool hi  = (lane >= 16);             // half-wave select (rows 2/3)

    v2f a; a.x = 1.0f; a.y = 1.0f;             // A = ones(16x4)
    v8f acc0 = {}; v8f acc1 = {}; v8f acc2 = {}; v8f acc3 = {};

    for (int e = beg; e < end; e += 4) {
        // branchless tail: clamp entry positions (uniform s_min), always load,
        // scale invalid rows by 0.  Everything here is SGPR-resident.
        const int p1 = (e + 1 < end) ? e + 1 : elast;
        const int p2 = (e + 2 < end) ? e + 2 : elast;
        const int p3 = (e + 3 < end) ? e + 3 : elast;
        const int i0 = __builtin_amdgcn_readfirstlane(idx[e]);
        const int i1 = __builtin_amdgcn_readfirstlane(idx[p1]);
        const int i2 = __builtin_amdgcn_readfirstlane(idx[p2]);
        const int i3 = __builtin_amdgcn_readfirstlane(idx[p3]);
        const float m1 = (e + 1 < end) ? 1.0f : 0.0f;
        const float m2 = (e + 2 < end) ? 1.0f : 0.0f;
        const float m3 = (e + 3 < end) ? 1.0f : 0.0f;

        // prefetch next K-step's rows (global_prefetch_b8): 4 rows x 256B,
        // lanes 0-7 -> row e+4, ..., lanes 24-31 -> row e+7, 32B apart
        if (e + 4 < end) {                      // uniform branch
            const int pe = e + 4 + (lane >> 3);
            const int pf = idx[(pe < end) ? pe : elast];
            __builtin_prefetch(src + (size_t)pf * NFILT + (lane & 7) * 8, 0, 1);
        }

        // B operand source rows for this lane (scalar bases, lane-selected)
        const float* r02 = src + (size_t)(hi ? i2 : i0) * NFILT + col; // -> b.x
        const float* r13 = src + (size_t)(hi ? i3 : i1) * NFILT + col; // -> b.y
        const float  s02 = hi ? m2 : 1.0f;
        const float  s13 = hi ? m3 : m1;

        v2f b0, b1, b2, b3;
        b0.x = r02[ 0] * s02;  b0.y = r13[ 0] * s13;
        b1.x = r02[16] * s02;  b1.y = r13[16] * s13;
        b2.x = r02[32] * s02;  b2.y = r13[32] * s13;
        b3.x = r02[48] * s02;  b3.y = r13[48] * s13;

        acc0 = __builtin_amdgcn_wmma_f32_16x16x4_f32(false, a, false, b0,
                                                     (short)0, acc0, false, false);
        acc1 = __builtin_amdgcn_wmma_f32_16x16x4_f32(false, a, false, b1,
                                                     (short)0, acc1, false, false);
        acc2 = __builtin_amdgcn_wmma_f32_16x16x4_f32(false, a, false, b2,
                                                     (short)0, acc2, false, false);
        acc3 = __builtin_amdgcn_wmma_f32_16x16x4_f32(false, a, false, b3,
                                                     (short)0, acc3, false, false);
    }

    // Every row of each D tile equals the column sum, so element 0 (VGPR0)
    // holds the sum for column `col` of that chunk in every lane.
    const float inv = 1.0f / (float)((cnt > 0) ? cnt : 1);
    const float o_lo = (hi ? acc1[0] : acc0[0]) * inv;  // chunk 1 : chunk 0
    const float o_hi = (hi ? acc3[0] : acc2[0]) * inv;  // chunk 3 : chunk 2
    const int   c_lo = hi ? 1 : 0;
    const int   c_hi = hi ? 3 : 2;

    float* orow = out + (size_t)t * NFILT;
    orow[c_lo * 16 + col] = o_lo;
    orow[c_hi * 16 + col] = o_hi;
}

extern "C" void kernel_launch(void* const* d_in, const int* in_sizes, int n_in,
                              void* d_out, int out_size, void* d_ws, size_t ws_size,
                              hipStream_t stream) {
    (void)n_in; (void)out_size; (void)d_ws; (void)ws_size;
    const float* sources = (const float*)d_in[0];
    const int*   indices = (const int*)d_in[1];
    const int*   spans   = (const int*)d_in[2];
    float*       out     = (float*)d_out;

    const int n_tgt = in_sizes[2] - 1;   // spans has N_TGT+1 entries
    const int blocks = (n_tgt + WAVES_PER_BLOCK - 1) / WAVES_PER_BLOCK;

    seg_mean_wmma_kernel<<<blocks, 256, 0, stream>>>(sources, indices, spans,
                                                     out, n_tgt);
}